// OrthSGNN_7301444403250
// MI455X (gfx1250) — compile-verified
//
#include <hip/hip_runtime.h>
#include <hip/hip_bf16.h>
#include <stdint.h>
#include <stddef.h>

// Problem constants (match reference)
#define N_NODES 100000
#define E_EDGES 3200000
#define F_DIM   512
#define C_OUT   64
#define K_ORD   10

typedef float v2f __attribute__((ext_vector_type(2)));
typedef float v8f __attribute__((ext_vector_type(8)));

// ---------------- utility kernels ----------------

__global__ __launch_bounds__(256) void zero_kernel(float* __restrict__ p, int n) {
    int i = blockIdx.x * 256 + threadIdx.x;
    if (i < n) p[i] = 0.0f;
}

__global__ __launch_bounds__(256) void deg_kernel(const int* __restrict__ col,
                                                  float* __restrict__ deg, int e) {
    int i = blockIdx.x * 256 + threadIdx.x;
    if (i < e) atomicAdd(&deg[col[i]], 1.0f);
}

__global__ __launch_bounds__(256) void dinv_kernel(float* __restrict__ d, int n) {
    int i = blockIdx.x * 256 + threadIdx.x;
    if (i < n) {
        float v = d[i];
        d[i] = (v > 0.0f) ? rsqrtf(fmaxf(v, 1.0f)) : 1.0f;
    }
}

__global__ __launch_bounds__(256) void norm_kernel(const int* __restrict__ row,
                                                   const int* __restrict__ col,
                                                   const float* __restrict__ dinv,
                                                   float* __restrict__ nrm, int e) {
    int i = blockIdx.x * 256 + threadIdx.x;
    if (i < e) nrm[i] = dinv[row[i]] * dinv[col[i]];
}

// ---------------- WMMA GEMM: h = x @ W + b ----------------
// x: [N, 512] f32, W: [512, 64] f32, b: [64]
// One wave computes a 16x64 output strip (four 16x16 C tiles).
// W is staged through LDS TRANSPOSED ([c][k], column stride 132 ≡ 4 mod 64)
// so each B fragment (W[k][c], W[k+1][c]) is one conflict-free ds_load_b64
// landing directly in an even-aligned VGPR pair for WMMA.
#define K_CHUNK    128
#define LDS_CSTR   132   // per-column K stride (floats); 132 mod 64 == 4

__global__ __launch_bounds__(256) void gemm_kernel(const float* __restrict__ x,
                                                   const float* __restrict__ W,
                                                   const float* __restrict__ b,
                                                   float* __restrict__ h) {
    __shared__ float ldsWT[C_OUT * LDS_CSTR];  // 33 KB, transposed W chunk

    const int tid  = threadIdx.x;
    const int wave = tid >> 5;
    const int lane = tid & 31;

    const int rowTile = blockIdx.x * 8 + wave;          // 16-row tile index
    const bool active = (rowTile < (N_NODES / 16));     // 6250 tiles exactly

    const int m    = lane & 15;          // M (and N) index within tile
    const int koff = (lane >> 4) << 1;   // 0 for lanes 0-15, 2 for lanes 16-31
    const int r0   = rowTile * 16;
    const float* xrow = x + (size_t)(r0 + m) * F_DIM;

    v8f acc[4];
    #pragma unroll
    for (int t = 0; t < 4; ++t) {
        float bv = active ? b[t * 16 + m] : 0.0f;
        #pragma unroll
        for (int r = 0; r < 8; ++r) acc[t][r] = bv;
    }

    for (int kb = 0; kb < F_DIM; kb += K_CHUNK) {
        __syncthreads();
        // cooperative transpose-load of W[kb .. kb+127][0..63] into LDS
        // global: consecutive tid -> consecutive c (coalesced 256B per 64 lanes)
        for (int idx = tid; idx < K_CHUNK * C_OUT; idx += 256) {
            int kk = idx >> 6;
            int c  = idx & 63;
            ldsWT[c * LDS_CSTR + kk] = W[(size_t)(kb + kk) * C_OUT + c];
        }
        __syncthreads();

        if (active) {
            #pragma unroll 4
            for (int kk = 0; kk < K_CHUNK; kk += 4) {
                // A fragment: 16x4 f32. lane -> row m, VGPR0/1 = K koff / koff+1
                v2f a = *(const v2f*)(xrow + kb + kk + koff);
                #pragma unroll
                for (int t = 0; t < 4; ++t) {
                    // B fragment: 4x16 f32. lane -> col (t*16+m), rows koff/koff+1
                    // transposed LDS: contiguous pair -> single ds_load_b64
                    v2f bb = *(const v2f*)(&ldsWT[(t * 16 + m) * LDS_CSTR + kk + koff]);
                    acc[t] = __builtin_amdgcn_wmma_f32_16x16x4_f32(
                        /*neg_a=*/false, a, /*neg_b=*/false, bb,
                        /*c_mod=*/(short)0, acc[t],
                        /*reuse_a=*/false, /*reuse_b=*/false);
                }
            }
        }
    }

    if (active) {
        // C/D layout: VGPR r, lanes 0-15 -> M=r, lanes 16-31 -> M=r+8; N = lane&15
        const int rowAdd = (lane >> 4) * 8;
        float* hbase = h + (size_t)r0 * C_OUT;
        #pragma unroll
        for (int t = 0; t < 4; ++t)
            #pragma unroll
            for (int r = 0; r < 8; ++r)
                hbase[(size_t)(r + rowAdd) * C_OUT + t * 16 + m] = acc[t][r];
    }
}

// ---------------- edge scatter: out[col] += norm * in[row] ----------------
// One wave per edge; lane covers 2 consecutive channels -> fully coalesced
// 256B gather + same-cacheline f32 atomics.
__global__ __launch_bounds__(256) void prop_scatter(const float* __restrict__ in,
                                                    float* __restrict__ out,
                                                    const int* __restrict__ row,
                                                    const int* __restrict__ col,
                                                    const float* __restrict__ nrm,
                                                    int e) {
    int g    = blockIdx.x * 256 + threadIdx.x;
    int edge = g >> 5;
    int lane = threadIdx.x & 31;
    if (edge >= e) return;
    int   r  = row[edge];
    int   c  = col[edge];
    float nm = nrm[edge];
    v2f v = *(const v2f*)(in + (size_t)r * C_OUT + lane * 2);
    float* dst = out + (size_t)c * C_OUT + lane * 2;
    atomicAdd(dst,     nm * v.x);
    atomicAdd(dst + 1, nm * v.y);
}

// ---------------- ret = mf[0]⊙P0 + lap[0]*mf[1]⊙P1 ----------------
__global__ __launch_bounds__(256) void init_ret_kernel(const float* __restrict__ P0,
                                                       const float* __restrict__ P1,
                                                       const float* __restrict__ lap,
                                                       const float* __restrict__ mf,
                                                       float* __restrict__ ret, int n) {
    int i = blockIdx.x * 256 + threadIdx.x;
    if (i >= n) return;
    int c = i & 63;
    ret[i] = mf[c] * P0[i] + lap[0] * mf[C_OUT + c] * P1[i];
}

// ---------------- Pn = 2*T - Pprev2 (in place into Pprev2); ret += coef⊙Pn ----
__global__ __launch_bounds__(256) void combine_kernel(const float* __restrict__ T,
                                                      float* __restrict__ Pp2,
                                                      const float* __restrict__ lap,
                                                      const float* __restrict__ mf,
                                                      float* __restrict__ ret,
                                                      int order, int n) {
    int i = blockIdx.x * 256 + threadIdx.x;
    if (i >= n) return;
    int c = i & 63;
    float pn = 2.0f * T[i] - Pp2[i];
    Pp2[i] = pn;
    ret[i] += lap[order - 1] * mf[(size_t)order * C_OUT + c] * pn;
}

// ---------------- log_softmax over 64 channels, wave per node ----------------
__global__ __launch_bounds__(256) void lsm_kernel(const float* __restrict__ ret,
                                                  float* __restrict__ out, int n) {
    int g    = blockIdx.x * 256 + threadIdx.x;
    int node = g >> 5;
    int lane = threadIdx.x & 31;
    if (node >= n) return;
    const float* rrow = ret + (size_t)node * C_OUT;
    v2f v = *(const v2f*)(rrow + lane * 2);
    float m = fmaxf(v.x, v.y);
    #pragma unroll
    for (int off = 16; off > 0; off >>= 1)
        m = fmaxf(m, __shfl_xor(m, off, 32));
    float s = expf(v.x - m) + expf(v.y - m);
    #pragma unroll
    for (int off = 16; off > 0; off >>= 1)
        s += __shfl_xor(s, off, 32);
    float lse = m + logf(s);
    v2f o; o.x = v.x - lse; o.y = v.y - lse;
    *(v2f*)(out + (size_t)node * C_OUT + lane * 2) = o;
}

// ---------------- host orchestration ----------------

static inline size_t align256(size_t x) { return (x + 255) & ~(size_t)255; }

extern "C" void kernel_launch(void* const* d_in, const int* in_sizes, int n_in,
                              void* d_out, int out_size, void* d_ws, size_t ws_size,
                              hipStream_t stream) {
    const float* x   = (const float*)d_in[0];
    const int*   ei  = (const int*)d_in[1];      // [2, E] int32
    const float* W   = (const float*)d_in[2];
    const float* b   = (const float*)d_in[3];
    const float* lap = (const float*)d_in[4];
    const float* mf  = (const float*)d_in[5];
    float* out = (float*)d_out;

    const int*   row = ei;
    const int*   col = ei + E_EDGES;

    // workspace layout
    char* w = (char*)d_ws;
    float* dinv = (float*)w;  w += align256((size_t)N_NODES * sizeof(float));
    float* nrm  = (float*)w;  w += align256((size_t)E_EDGES * sizeof(float));
    const size_t NC = (size_t)N_NODES * C_OUT;
    float* bufA = (float*)w;  w += align256(NC * sizeof(float));  // P_{n-2}
    float* bufB = (float*)w;  w += align256(NC * sizeof(float));  // P_{n-1}
    float* bufT = (float*)w;  w += align256(NC * sizeof(float));  // prop scratch
    float* ret  = (float*)w;  w += align256(NC * sizeof(float));

    const int nNC = (int)NC;
    dim3 blk(256);
    dim3 gN((N_NODES + 255) / 256);
    dim3 gE((E_EDGES + 255) / 256);
    dim3 gNC((nNC + 255) / 256);
    dim3 gGemm((N_NODES / 16 + 7) / 8);                 // 782 blocks, 8 waves each
    dim3 gProp((int)(((size_t)E_EDGES * 32 + 255) / 256));  // wave per edge
    dim3 gLsm((int)(((size_t)N_NODES * 32 + 255) / 256));   // wave per node

    // 1) degree -> dinv
    zero_kernel<<<gN, blk, 0, stream>>>(dinv, N_NODES);
    deg_kernel<<<gE, blk, 0, stream>>>(col, dinv, E_EDGES);
    dinv_kernel<<<gN, blk, 0, stream>>>(dinv, N_NODES);
    norm_kernel<<<gE, blk, 0, stream>>>(row, col, dinv, nrm, E_EDGES);

    // 2) h = x @ W + b  (P0 in bufA)
    gemm_kernel<<<gGemm, blk, 0, stream>>>(x, W, b, bufA);

    // 3) P1 = prop(P0) in bufB
    zero_kernel<<<gNC, blk, 0, stream>>>(bufB, nNC);
    prop_scatter<<<gProp, blk, 0, stream>>>(bufA, bufB, row, col, nrm, E_EDGES);

    // 4) ret = mf0⊙P0 + lap0*mf1⊙P1
    init_ret_kernel<<<gNC, blk, 0, stream>>>(bufA, bufB, lap, mf, ret, nNC);

    // 5) Chebyshev recurrence, n = 2..K
    float* Pp2 = bufA;   // P_{n-2}
    float* Pp1 = bufB;   // P_{n-1}
    for (int n = 2; n <= K_ORD; ++n) {
        zero_kernel<<<gNC, blk, 0, stream>>>(bufT, nNC);
        prop_scatter<<<gProp, blk, 0, stream>>>(Pp1, bufT, row, col, nrm, E_EDGES);
        combine_kernel<<<gNC, blk, 0, stream>>>(bufT, Pp2, lap, mf, ret, n, nNC);
        // Pp2 now holds P_n
        float* t = Pp1; Pp1 = Pp2; Pp2 = t;
    }

    // 6) log_softmax
    lsm_kernel<<<gLsm, blk, 0, stream>>>(ret, out, N_NODES);

    (void)in_sizes; (void)n_in; (void)out_size; (void)ws_size;
}